// WeightAndSumThenCat_77635828843232
// MI455X (gfx1250) — compile-verified
//
#include <hip/hip_runtime.h>
#include <stdint.h>

// ============================================================================
// WeightAndSumThenCat on MI455X (gfx1250).
//
// HBM-bound op (~1 GB streamed, ~1 GFLOP): stream each feature row once via
// gfx1250 async global->LDS b128 loads, compute 16 gate dot-products per tile
// with V_WMMA_F32_16X16X4_F32 (B = W broadcast over N), gate-weight and
// segment-accumulate in registers (batch sorted), flush per-graph partials
// with native f32 global atomics.
// ============================================================================

typedef float v2f __attribute__((ext_vector_type(2)));
typedef float v8f __attribute__((ext_vector_type(8)));

#define OUT_STRIDE 448

__device__ __forceinline__ uint32_t lds_lo32(const void* p) {
  // generic pointer to LDS: addr[31:0] is the LDS byte offset (ISA 10.2)
  return (uint32_t)(uintptr_t)p;
}

__device__ __forceinline__ void async_load_b128(uint32_t lds_byte, const float* g) {
  asm volatile("global_load_async_to_lds_b128 %0, %1, off"
               :: "v"(lds_byte), "v"((uint64_t)(uintptr_t)g)
               : "memory");
}
__device__ __forceinline__ void wait_async_0() {
  asm volatile("s_wait_asynccnt 0" ::: "memory");
}
__device__ __forceinline__ void wait_ds_0() {
  asm volatile("s_wait_dscnt 0" ::: "memory");
}

__device__ __forceinline__ void atomic_add_f32(float* p, float v) {
  unsafeAtomicAdd(p, v);  // native global_atomic_add_f32
}

template <int D, int COLBASE, int ROWS_PER_WAVE>
__global__ __launch_bounds__(64)
void pool_kernel(const float* __restrict__ feats,
                 const float* __restrict__ W,
                 const float* __restrict__ bias,
                 const int*   __restrict__ batch,
                 float* __restrict__ out,
                 int n_rows)
{
  constexpr int S   = D + 4;        // LDS row stride (floats); S%64==4 -> no bank conflicts
  constexpr int CPL = D / 32;       // columns owned per lane
  constexpr int F4_PER_ROW   = D / 4;
  constexpr int F4_PER_LANE  = (16 * F4_PER_ROW) / 32;

  __shared__ float sW[D];
  __shared__ float sT[2][16 * S];

  const int tid  = threadIdx.x;
  const int wave = tid >> 5;
  const int lane = tid & 31;

  for (int d = tid; d < D; d += 64) sW[d] = W[d];
  __syncthreads();

  const long wave_id   = (long)blockIdx.x * 2 + wave;
  const long row_begin = wave_id * (long)ROWS_PER_WAVE;
  if (row_begin >= n_rows) return;           // wave-uniform exit

  const float b0 = bias[0];
  float* sm = sT[wave];
  const uint32_t smbase = lds_lo32(sm);

  const int m = lane & 15;  // WMMA A row fed by this lane
  const int h = lane >> 4;  // half-wave selects K pair {0,1} vs {2,3}

  float acc[CPL];
#pragma unroll
  for (int j = 0; j < CPL; ++j) acc[j] = 0.0f;
  int cur_g = -1;

  for (int c0 = 0; c0 < ROWS_PER_WAVE; c0 += 16) {
    const long row0 = row_begin + c0;
    if (row0 >= n_rows) break;               // wave-uniform
    const int nv = (n_rows - row0 < 16) ? (int)(n_rows - row0) : 16;

    // previous tile's LDS reads must retire before async writes overwrite it
    wait_ds_0();

    // ---- stage 16 rows into LDS, coalesced 16B/lane async loads ----
#pragma unroll
    for (int i = 0; i < F4_PER_LANE; ++i) {
      const int linear = i * 32 + lane;
      const int r = linear / F4_PER_ROW;     // constexpr pow2 -> shift
      const int f = linear % F4_PER_ROW;
      if (r < nv) {
        async_load_b128(smbase + (uint32_t)((r * S + f * 4) * 4),
                        feats + (row0 + r) * (long)D + f * 4);
      }
    }

    // sorted segment id for row0 + (lane&15)
    long bi = row0 + m;
    if (bi >= n_rows) bi = n_rows - 1;
    const int mybatch = batch[bi];

    wait_async_0();

    // ---- 16 gate dot-products ----
    float dots[16];
#if __has_builtin(__builtin_amdgcn_wmma_f32_16x16x4_f32)
    // A[m][k]=feats[row m][k0+k] (ISA 16x4 f32 layout), B[k][n]=W[k0+k] bcast.
    v8f cacc = {0.f,0.f,0.f,0.f,0.f,0.f,0.f,0.f};
#pragma unroll 8
    for (int k0 = 0; k0 < D; k0 += 4) {
      const v2f a = *(const v2f*)(sm + m * S + k0 + 2 * h);
      const v2f b = *(const v2f*)(sW + k0 + 2 * h);
      cacc = __builtin_amdgcn_wmma_f32_16x16x4_f32(
          false, a, false, b, (short)0, cacc, false, false);
    }
    // D layout: VGPR j holds row (j + 8*half), identical across the N lanes
#pragma unroll
    for (int r = 0; r < 16; ++r) {
      dots[r] = __int_as_float(__builtin_amdgcn_readlane(
          __float_as_int(cacc[r & 7]), (r & 8) ? 16 : 0));
    }
#else
    // fallback: per-lane partial dots + xor-shuffle reduction
#pragma unroll
    for (int r = 0; r < 16; ++r) {
      float s = 0.0f;
#pragma unroll
      for (int j = 0; j < CPL; ++j)
        s += sm[r * S + lane + 32 * j] * sW[lane + 32 * j];
#pragma unroll
      for (int off = 16; off > 0; off >>= 1) s += __shfl_xor(s, off, 32);
      dots[r] = s;
    }
#endif

    // ---- gate + segment accumulate (graph id uniform per row) ----
#pragma unroll
    for (int r = 0; r < 16; ++r) {
      if (r < nv) {                          // wave-uniform
        const float x    = dots[r] + b0;
        const float gate = 1.0f / (1.0f + __expf(-x));
        const int   g    = __builtin_amdgcn_readlane(mybatch, r);
        if (g != cur_g) {                    // wave-uniform (SGPR compare)
          if (cur_g >= 0) {
            float* dst = out + (long)cur_g * OUT_STRIDE + COLBASE + lane;
#pragma unroll
            for (int j = 0; j < CPL; ++j) {
              atomic_add_f32(dst + 32 * j, acc[j]);
              acc[j] = 0.0f;
            }
          }
          cur_g = g;
        }
#pragma unroll
        for (int j = 0; j < CPL; ++j)
          acc[j] = fmaf(gate, sm[r * S + lane + 32 * j], acc[j]);
      }
    }
  }

  if (cur_g >= 0) {
    float* dst = out + (long)cur_g * OUT_STRIDE + COLBASE + lane;
#pragma unroll
    for (int j = 0; j < CPL; ++j) atomic_add_f32(dst + 32 * j, acc[j]);
  }
}

// zero pooled region, direct-copy global feats into cols [384,448)
__global__ void init_out_kernel(const float* __restrict__ gfeat,
                                float* __restrict__ out, int total)
{
  const int i = blockIdx.x * blockDim.x + threadIdx.x;
  if (i >= total) return;
  const int g = i / OUT_STRIDE;
  const int c = i - g * OUT_STRIDE;
  out[i] = (c < 384) ? 0.0f : gfeat[g * 64 + (c - 384)];
}

extern "C" void kernel_launch(void* const* d_in, const int* in_sizes, int n_in,
                              void* d_out, int out_size, void* d_ws, size_t ws_size,
                              hipStream_t stream)
{
  const float* atom_feats   = (const float*)d_in[0];
  const float* bond_feats   = (const float*)d_in[1];
  const float* global_feats = (const float*)d_in[2];
  const float* W_atom       = (const float*)d_in[3];
  const float* b_atom       = (const float*)d_in[4];
  const float* W_bond       = (const float*)d_in[5];
  const float* b_bond       = (const float*)d_in[6];
  const int*   atom_batch   = (const int*)d_in[7];
  const int*   bond_batch   = (const int*)d_in[8];
  float* out = (float*)d_out;

  const int n_atom = in_sizes[7];
  const int n_bond = in_sizes[8];

  // 1) init output (accumulators must start at zero; harness poisons d_out)
  {
    const int threads = 256;
    const int blocks  = (out_size + threads - 1) / threads;
    init_out_kernel<<<blocks, threads, 0, stream>>>(global_feats, out, out_size);
  }

  constexpr int RPW = 128;  // rows per wave: ~2 segment flushes per wave
  // 2) atoms: D=256 -> out cols [0,256)
  {
    const long waves  = ((long)n_atom + RPW - 1) / RPW;
    const int  blocks = (int)((waves + 1) / 2);
    pool_kernel<256, 0, RPW><<<blocks, 64, 0, stream>>>(
        atom_feats, W_atom, b_atom, atom_batch, out, n_atom);
  }
  // 3) bonds: D=128 -> out cols [256,384)
  {
    const long waves  = ((long)n_bond + RPW - 1) / RPW;
    const int  blocks = (int)((waves + 1) / 2);
    pool_kernel<128, 256, RPW><<<blocks, 64, 0, stream>>>(
        bond_feats, W_bond, b_bond, bond_batch, out, n_bond);
  }
  (void)n_in; (void)d_ws; (void)ws_size;
}